// CheckLayer_71614284693527
// MI455X (gfx1250) — compile-verified
//
#include <hip/hip_runtime.h>
#include <hip/hip_bf16.h>
#include <stdint.h>

#define N_NODES 65536
#define NB      64          // batch
#define MAXNBR  32
#define BIGF    1e10f
#define EPSF    1e-10f
#define TSTRIDE 65          // 64 + 1 padding -> conflict-free strided LDS reads

// ---------------------------------------------------------------------------
// CDNA5 async global->LDS helpers (ASYNCcnt path, CDNA5 ISA)
// saddr form: global addr = SGPR64 base + VGPR32 byte offset (GVS mode).
// ---------------------------------------------------------------------------
__device__ __forceinline__ void async_g2lds_b32_saddr(uint32_t lds_byte_off,
                                                      uint32_t g_byte_off,
                                                      const float* base) {
    asm volatile("global_load_async_to_lds_b32 %0, %1, %2"
                 :: "v"(lds_byte_off), "v"(g_byte_off), "s"(base) : "memory");
}
__device__ __forceinline__ void wait_asynccnt0() {
    asm volatile("s_wait_asynccnt 0" ::: "memory");
}

// ---------------------------------------------------------------------------
// Kernel A: transpose input [64][65536] -> inT [65537][64]; row 65536 = zeros.
// 64(node) x 64(batch) tile per 256-thread block, staged in LDS via async
// global->LDS loads (no VGPR round trip), then written out coalesced.
// ---------------------------------------------------------------------------
__global__ void __launch_bounds__(256)
transpose_in_kernel(const float* __restrict__ in, float* __restrict__ inT) {
    __shared__ float tile[64 * TSTRIDE];

    const int t  = threadIdx.x;       // 0..255
    const int n0 = blockIdx.x * 64;
    const int nl = t & 63;            // node-local, consecutive across lanes -> coalesced
    const int bq = t >> 6;            // 0..3

    const uint32_t l0 = (uint32_t)(size_t)(void*)&tile[0];
    // Load phase: 16 async b32 per thread; lane addresses consecutive in both
    // global memory (128B per wave) and LDS (conflict-free).
    #pragma unroll
    for (int r = 0; r < 16; ++r) {
        const int b = bq + r * 4;     // covers b = 0..63 exactly once
        const uint32_t loff = l0 + (uint32_t)(b * TSTRIDE + nl) * 4u;
        const uint32_t goff = ((uint32_t)b * (uint32_t)N_NODES
                               + (uint32_t)(n0 + nl)) * 4u;   // < 16 MB, fits u32
        async_g2lds_b32_saddr(loff, goff, in);
    }
    wait_asynccnt0();                  // this wave's async writes to LDS done
    __syncthreads();                   // all waves' writes visible

    // Store phase: lanes sweep batch quads -> 256B contiguous stores per 16 lanes.
    const int b4    = (t & 15) * 4;    // batch quad
    const int nbase = t >> 4;          // 0..15
    #pragma unroll
    for (int r = 0; r < 4; ++r) {
        const int n = nbase + r * 16;
        float4 v;
        v.x = tile[(b4 + 0) * TSTRIDE + n];   // stride 65 floats -> distinct banks
        v.y = tile[(b4 + 1) * TSTRIDE + n];
        v.z = tile[(b4 + 2) * TSTRIDE + n];
        v.w = tile[(b4 + 3) * TSTRIDE + n];
        *(float4*)(inT + (size_t)(n0 + n) * NB + b4) = v;
    }

    // Zero row N_NODES: branchless target for idx == -1.
    if (blockIdx.x == 0 && t < NB) {
        inT[(size_t)N_NODES * NB + t] = 0.0f;
    }
}

// ---------------------------------------------------------------------------
// Kernel B: gather + reduce + transposed write-out.
// Block = 256 threads = 64 batches x 4 node-teams; block covers 64 nodes.
// Per node: 32 scalar-loaded indices (wave-uniform via readfirstlane),
// 32 coalesced gathers from inT[node][0..63].
// Sign product kept exact via XOR of sign bits + zero flag.
// ---------------------------------------------------------------------------
__global__ void __launch_bounds__(256)
gather_reduce_kernel(const float* __restrict__ inT, const int* __restrict__ idx,
                     float* __restrict__ out) {
    __shared__ float res[64 * TSTRIDE];

    const int t    = threadIdx.x;
    const int n0   = blockIdx.x * 64;
    const int b    = t & 63;           // batch lane
    const int team = t >> 6;           // 0..3 (uniform within each wave)

    for (int nl = team; nl < 64; nl += 4) {
        // node is uniform per wave -> force SGPR so indices use scalar loads
        const int node = __builtin_amdgcn_readfirstlane(n0 + nl);
        const int* ip  = idx + (size_t)node * MAXNBR;

        uint32_t flip = 0u;            // XOR of sign bits of (f + eps)
        uint32_t zero = 0u;            // saw (f + eps) == 0 exactly
        float    m    = BIGF;
        #pragma unroll
        for (int k = 0; k < MAXNBR; ++k) {
            const int j    = ip[k];                       // s_load (uniform)
            const int safe = (j < 0) ? N_NODES : j;       // zero row if invalid
            const float f  = inT[(size_t)safe * NB + b];  // coalesced 128B/wave
            const float tv = f + EPSF;
            flip ^= __float_as_uint(tv);                  // low bits masked later
            zero |= (tv == 0.0f) ? 1u : 0u;
            float a = fabsf(f);                           // folds to |src| modifier
            a = (a == 0.0f) ? BIGF : a;
            m = fminf(m, a);                              // exact: min
        }
        const float sgn = zero ? 0.0f
                               : __uint_as_float(0x3F800000u | (flip & 0x80000000u));
        res[nl * TSTRIDE + b] = sgn * m;                  // conflict-free row write
    }
    __syncthreads();

    // Write-out transposed: out[b][n], lanes sweep n -> 256B contiguous stores.
    const int nc4 = (t & 15) * 4;
    #pragma unroll
    for (int r = 0; r < 4; ++r) {
        const int bo = (t >> 4) + r * 16;   // 0..63
        float4 v;
        v.x = res[(nc4 + 0) * TSTRIDE + bo];
        v.y = res[(nc4 + 1) * TSTRIDE + bo];
        v.z = res[(nc4 + 2) * TSTRIDE + bo];
        v.w = res[(nc4 + 3) * TSTRIDE + bo];
        *(float4*)(out + (size_t)bo * N_NODES + (size_t)(n0 + nc4)) = v;
    }
}

// ---------------------------------------------------------------------------
// Fallback (only if d_ws is too small for the 16.8 MB transposed input):
// correct but uncoalesced direct gather.
// ---------------------------------------------------------------------------
__global__ void __launch_bounds__(64)
direct_kernel(const float* __restrict__ in, const int* __restrict__ idx,
              float* __restrict__ out) {
    const int n = blockIdx.x;
    const int b = threadIdx.x;        // 0..63
    const int* ip = idx + (size_t)n * MAXNBR;

    uint32_t flip = 0u, zero = 0u;
    float m = BIGF;
    #pragma unroll
    for (int k = 0; k < MAXNBR; ++k) {
        const int j = ip[k];
        if (j >= 0) {                  // wave-uniform branch (j is scalar)
            const float f  = in[(size_t)b * N_NODES + j];
            const float tv = f + EPSF;
            flip ^= __float_as_uint(tv);
            zero |= (tv == 0.0f) ? 1u : 0u;
            float a = fabsf(f);
            a = (a == 0.0f) ? BIGF : a;
            m = fminf(m, a);
        }
    }
    const float sgn = zero ? 0.0f
                           : __uint_as_float(0x3F800000u | (flip & 0x80000000u));
    out[(size_t)b * N_NODES + n] = sgn * m;
}

// ---------------------------------------------------------------------------
extern "C" void kernel_launch(void* const* d_in, const int* in_sizes, int n_in,
                              void* d_out, int out_size, void* d_ws, size_t ws_size,
                              hipStream_t stream) {
    const float* in  = (const float*)d_in[0];   // [64][65536] fp32
    const int*   idx = (const int*)d_in[1];     // [65536][32] int32
    float*       out = (float*)d_out;           // [64][65536] fp32

    const size_t need = (size_t)(N_NODES + 1) * NB * sizeof(float);
    if (ws_size >= need) {
        float* inT = (float*)d_ws;              // [65537][64] fp32
        transpose_in_kernel<<<N_NODES / 64, 256, 0, stream>>>(in, inT);
        gather_reduce_kernel<<<N_NODES / 64, 256, 0, stream>>>(inT, idx, out);
    } else {
        direct_kernel<<<N_NODES, 64, 0, stream>>>(in, idx, out);
    }
}